// SparseMemory_50362786513310
// MI455X (gfx1250) — compile-verified
//
#include <hip/hip_runtime.h>
#include <stdint.h>

typedef __attribute__((ext_vector_type(2))) float v2f;
typedef __attribute__((ext_vector_type(8))) float v8f;

#define B_    32
#define M_    65536
#define W_    32
#define R_    4
#define DIN   256
#define KNB   8
#define SLABS 64
#define MSLAB (M_ / SLABS)   // 1024 columns per workgroup
#define TILES (MSLAB / 16)   // 64 WMMA tiles per workgroup

// ---- workspace layout (bytes) ----
// [0      , 16384 )  keysNeg2 : float[32][128]   (-2 * read_keys)
// [16384  , 16896 )  knorm    : float[32][4]     (||read_keys||^2)
// [32768  , 557056)  cand     : u64[32*4][SLABS][8] packed (orderedF32(score)<<32)|m
#define WS_KEYS  0
#define WS_KNORM 16384
#define WS_CAND  32768

__device__ __forceinline__ uint32_t orderF32(float f) {
  uint32_t u = __float_as_uint(f);
  return u ^ ((u >> 31) ? 0xFFFFFFFFu : 0x80000000u);
}
__device__ __forceinline__ float unorderF32(uint32_t v) {
  uint32_t u = v ^ ((v & 0x80000000u) ? 0x80000000u : 0xFFFFFFFFu);
  return __uint_as_float(u);
}

// insert candidate into ascending sorted-8 list (cheap reject on most calls)
#define INSERT8(lstarr, key)                                                  \
  do {                                                                        \
    if ((key) < (lstarr)[KNB - 1]) {                                          \
      (lstarr)[KNB - 1] = (key);                                              \
      _Pragma("unroll")                                                       \
      for (int _j = KNB - 1; _j > 0; --_j) {                                  \
        uint64_t _lo = (lstarr)[_j - 1], _hi = (lstarr)[_j];                  \
        bool _sw = _hi < _lo;                                                 \
        (lstarr)[_j - 1] = _sw ? _hi : _lo;                                   \
        (lstarr)[_j]     = _sw ? _lo : _hi;                                   \
      }                                                                       \
    }                                                                         \
  } while (0)

// ---------------- Kernel 0: read_keys = xi @ W^T + b  (store -2*rk and ||rk||^2)
__global__ void __launch_bounds__(128)
sm_keys_kernel(const float* __restrict__ xi, const float* __restrict__ W,
               const float* __restrict__ brk, float* __restrict__ keysNeg2,
               float* __restrict__ knorm) {
  const int b = blockIdx.x;
  const int j = threadIdx.x;               // 0..127  (r*32+w)
  __shared__ float s_rk[128];
  const float* xrow = xi + (size_t)b * DIN;
  const float* wrow = W + (size_t)j * DIN;
  float acc = brk[j];
#pragma unroll 8
  for (int d = 0; d < DIN; ++d) acc += xrow[d] * wrow[d];
  s_rk[j] = acc;
  keysNeg2[b * 128 + j] = -2.0f * acc;
  __syncthreads();
  if (j < R_) {
    float s = 0.0f;
#pragma unroll
    for (int w = 0; w < W_; ++w) { float v = s_rk[j * W_ + w]; s += v * v; }
    knorm[b * R_ + j] = s;
  }
}

// ---------------- Kernel 1: streaming WMMA scores + per-slab top-8 per (b,r)
__global__ void __launch_bounds__(32)
sm_topk_kernel(const float* __restrict__ sparse,
               const float* __restrict__ keysNeg2,
               uint64_t* __restrict__ cand) {
  const int slab = blockIdx.x % SLABS;
  const int b    = blockIdx.x / SLABS;
  const int lane = threadIdx.x;
  const int nsub = lane & 15;   // N (column-in-tile) for B/C/D, M (row) for A
  const int half = lane >> 4;   // K half-select

  // A fragment: A[M=r][K=w] = -2*rk[b][r][w]; rows 4..15 zero.
  v2f afrag[8];
  const float* kb = keysNeg2 + b * 128;
#pragma unroll
  for (int k = 0; k < 8; ++k) {
    const int w = 4 * k + 2 * half;
    float a0 = 0.0f, a1 = 0.0f;
    if (nsub < R_) { a0 = kb[nsub * W_ + w]; a1 = kb[nsub * W_ + w + 1]; }
    afrag[k].x = a0; afrag[k].y = a1;
  }

  uint64_t lst[R_][KNB];
#pragma unroll
  for (int r = 0; r < R_; ++r)
#pragma unroll
    for (int j = 0; j < KNB; ++j) lst[r][j] = ~0ull;

  const int m_base = slab * MSLAB;
  const float* sb = sparse + (size_t)b * M_ * W_;

  for (int t = 0; t < TILES; ++t) {
    const int m0  = m_base + t * 16;
    const int col = m0 + nsub;
    // lane's slice of the 32-float row: floats {4k+2*half, 4k+2*half+1}
    const v2f* p = (const v2f*)(sb + (size_t)col * W_ + 2 * half);
    v2f bfrag[8];
#pragma unroll
    for (int k = 0; k < 8; ++k)
      bfrag[k] = __builtin_nontemporal_load(p + 2 * k);  // stream: 256MB > L2

    float nl = 0.0f;
    v8f d = {};
#pragma unroll
    for (int k = 0; k < 8; ++k) {
      nl += bfrag[k].x * bfrag[k].x + bfrag[k].y * bfrag[k].y;
      // D += A(16x4 slice) * B(4x16 slice): accumulates -2*<rk, s> over K=32
      d = __builtin_amdgcn_wmma_f32_16x16x4_f32(false, afrag[k], false, bfrag[k],
                                                (short)0, d, false, false);
    }
    const float nrm = nl + __shfl_xor(nl, 16, 32);  // full ||s||^2 for this col

    const uint32_t idx = (uint32_t)col;
#pragma unroll
    for (int r = 0; r < R_; ++r) {
      const float score = d[r] + nrm;                 // d2 - ||rk||^2 (order-preserving)
      uint64_t key = ((uint64_t)orderF32(score) << 32) | (uint64_t)idx;
      if (half != 0) key = ~0ull;                     // lanes 16..31 hold rows 8..11 (unused)
      INSERT8(lst[r], key);
    }
  }

  // merge 16 lanes * 8 entries -> top-8 per r, write to workspace
  __shared__ uint64_t smem[16 * KNB];
  for (int r = 0; r < R_; ++r) {
    __syncthreads();
    if (half == 0) {
#pragma unroll
      for (int j = 0; j < KNB; ++j) smem[nsub * KNB + j] = lst[r][j];
    }
    __syncthreads();
    if (lane == 0) {
      uint64_t best[KNB];
#pragma unroll
      for (int j = 0; j < KNB; ++j) best[j] = ~0ull;
      for (int i = 0; i < 16 * KNB; ++i) {
        uint64_t c = smem[i];
        INSERT8(best, c);
      }
      uint64_t* dst = cand + (((size_t)(b * R_ + r)) * SLABS + slab) * KNB;
#pragma unroll
      for (int j = 0; j < KNB; ++j) dst[j] = best[j];
    }
  }
}

// ---------------- Kernel 2: merge slabs, decode, gather vectors, weights
__global__ void __launch_bounds__(32)
sm_final_kernel(const float* __restrict__ sparse, const uint64_t* __restrict__ cand,
                const float* __restrict__ knorm, float* __restrict__ out) {
  const int pair = blockIdx.x;       // b*4 + r
  const int b = pair >> 2, r = pair & 3;
  const int lane = threadIdx.x;

  const uint64_t* c = cand + (size_t)pair * SLABS * KNB;  // 512 sorted-ish candidates
  uint64_t loc[KNB];
#pragma unroll
  for (int j = 0; j < KNB; ++j) loc[j] = ~0ull;
  for (int i = 0; i < (SLABS * KNB) / 32; ++i) {          // 16 per lane
    uint64_t v = c[lane * ((SLABS * KNB) / 32) + i];
    INSERT8(loc, v);
  }

  __shared__ uint64_t smem[32 * KNB];
  __shared__ uint64_t fin[KNB];
#pragma unroll
  for (int j = 0; j < KNB; ++j) smem[lane * KNB + j] = loc[j];
  __syncthreads();
  if (lane == 0) {
    uint64_t best[KNB];
#pragma unroll
    for (int j = 0; j < KNB; ++j) best[j] = ~0ull;
    for (int i = 0; i < 32 * KNB; ++i) {
      uint64_t v = smem[i];
      INSERT8(best, v);
    }
#pragma unroll
    for (int j = 0; j < KNB; ++j) fin[j] = best[j];
  }
  __syncthreads();

  // decode distances: d2 = score + ||rk||^2 ; list ascending -> max is dist[7]
  float dist[KNB];
  const float kn = knorm[pair];
#pragma unroll
  for (int j = 0; j < KNB; ++j)
    dist[j] = unorderF32((uint32_t)(fin[j] >> 32)) + kn;
  const float wmax = dist[KNB - 1];

  // read_weights[r][b][j] at offset 32768
  if (lane < KNB)
    out[(size_t)B_ * R_ * KNB * W_ + ((size_t)(r * B_ + b)) * KNB + lane] =
        dist[lane] / wmax;

  // read_vectors[r][b][j][w] : 8 coalesced 128B row copies (lane == w)
  const float* sb = sparse + (size_t)b * M_ * W_;
#pragma unroll
  for (int j = 0; j < KNB; ++j) {
    const uint32_t idx = (uint32_t)(fin[j] & 0xFFFFFFFFu);
    out[(((size_t)(r * B_ + b)) * KNB + j) * W_ + lane] =
        sb[(size_t)idx * W_ + lane];
  }
}

extern "C" void kernel_launch(void* const* d_in, const int* in_sizes, int n_in,
                              void* d_out, int out_size, void* d_ws, size_t ws_size,
                              hipStream_t stream) {
  const float* xi     = (const float*)d_in[0];
  const float* sparse = (const float*)d_in[1];
  const float* W_rk   = (const float*)d_in[2];
  const float* b_rk   = (const float*)d_in[3];

  float*    keysNeg2 = (float*)((char*)d_ws + WS_KEYS);
  float*    knorm    = (float*)((char*)d_ws + WS_KNORM);
  uint64_t* cand     = (uint64_t*)((char*)d_ws + WS_CAND);

  sm_keys_kernel<<<B_, 128, 0, stream>>>(xi, W_rk, b_rk, keysNeg2, knorm);
  sm_topk_kernel<<<B_ * SLABS, 32, 0, stream>>>(sparse, keysNeg2, cand);
  sm_final_kernel<<<B_ * R_, 32, 0, stream>>>(sparse, cand, knorm, (float*)d_out);
}